// MoEGate_14078902796920
// MI455X (gfx1250) — compile-verified
//
#include <hip/hip_runtime.h>
#include <hip/hip_bf16.h>
#include <math.h>

typedef __attribute__((ext_vector_type(2))) float v2f;
typedef __attribute__((ext_vector_type(8))) float v8f;

#define EXPERTS   64
#define TOPK_K    8
#define BK        128   // K-chunk staged in LDS
#define TM        16    // tokens per block (WMMA M)
#define XS_STRIDE (BK + 4)   // pad: conflict-free A reads, keeps 16B alignment
#define LG_STRIDE (EXPERTS + 1)

// 4 waves per block: wave w handles experts [16w, 16w+16) for the same 16 tokens.
__global__ __launch_bounds__(128) void moe_gate_kernel(
    const float* __restrict__ x,      // [N, D]
    const float* __restrict__ w,      // [E, D]
    float* __restrict__ out,          // [N*K weights][N*K indices][1 aux]
    float* __restrict__ ws,           // ws[0..63]=sum probs, ws[64..127]=counts
    int D, int N)
{
    __shared__ float xs[TM * XS_STRIDE];        // x tile 16 x BK
    __shared__ float lg[TM * LG_STRIDE];        // logits tile 16 x 64
    __shared__ float pi_sh[EXPERTS];
    __shared__ float cnt_sh[EXPERTS];

    const int tid  = threadIdx.x;
    const int lane = tid & 31;
    const int wave = tid >> 5;
    const int tok0 = blockIdx.x * TM;

    // WMMA f32 16x16x4 fragment addressing (wave32):
    //   A (16x4): lanes 0-15 -> M=lane, K = k0+{0,1}; lanes 16-31 -> M=lane-16, K = k0+{2,3}
    //   B (4x16): lanes 0-15 -> N=lane, K = k0+{0,1}; lanes 16-31 -> N=lane-16, K = k0+{2,3}
    const int fm    = lane & 15;
    const int koff  = (lane >> 4) * 2;
    const int ebase = wave * 16;
    const float* wrow = w + (size_t)(ebase + fm) * D;

    v8f acc = {};

    const int nchunk = D / BK;
    for (int c = 0; c < nchunk; ++c) {
        const int kbase = c * BK;
        // cooperative coalesced stage: 16 rows x 32 float4 = 512 float4, 4/thread
        #pragma unroll
        for (int i = 0; i < (TM * BK / 4) / 128; ++i) {
            int flat = tid + i * 128;
            int row  = flat >> 5;     // BK/4 = 32 float4 per row
            int c4   = flat & 31;
            float4 v = *(const float4*)(x + (size_t)(tok0 + row) * D + kbase + c4 * 4);
            *(float4*)(&xs[row * XS_STRIDE + c4 * 4]) = v;
        }
        __syncthreads();
        #pragma unroll
        for (int k0 = 0; k0 < BK; k0 += 4) {
            v2f a = *(const v2f*)(&xs[fm * XS_STRIDE + k0 + koff]);
            v2f b = *(const v2f*)(wrow + kbase + k0 + koff);
            // 8 args: (neg_a, A, neg_b, B, c_mod, C, reuse_a, reuse_b)
            acc = __builtin_amdgcn_wmma_f32_16x16x4_f32(
                false, a, false, b, (short)0, acc, false, false);
        }
        __syncthreads();
    }

    // Scatter logit tile: lane holds column n=fm, rows v + 8*(lane>>4), v=0..7
    #pragma unroll
    for (int v = 0; v < 8; ++v) {
        int m = v + 8 * (lane >> 4);
        lg[m * LG_STRIDE + ebase + fm] = acc[v];
    }
    if (tid < EXPERTS) { pi_sh[tid] = 0.0f; cnt_sh[tid] = 0.0f; }
    __syncthreads();

    // Softmax + top-8 per token; 16 token-threads, all data resident in LDS.
    if (tid < TM) {
        float* row = &lg[tid * LG_STRIDE];

        float mx = -3.4e38f;
        for (int e = 0; e < EXPERTS; ++e) mx = fmaxf(mx, row[e]);

        float s = 0.0f;
        for (int e = 0; e < EXPERTS; ++e) {
            float v = __expf(row[e] - mx);
            row[e] = v;
            s += v;
        }
        float inv = 1.0f / s;
        for (int e = 0; e < EXPERTS; ++e) {
            float p = row[e] * inv;
            row[e] = p;
            atomicAdd(&pi_sh[e], p);          // pi: mean-prob accumulator
        }

        float tw[TOPK_K];
        int   ti[TOPK_K];
        float wsum = 0.0f;
        for (int k = 0; k < TOPK_K; ++k) {
            float bm = -1.0f; int bi = 0;
            for (int e = 0; e < EXPERTS; ++e) {
                float v = row[e];
                if (v > bm) { bm = v; bi = e; }
            }
            tw[k] = bm; ti[k] = bi; wsum += bm;
            row[bi] = -2.0f;
            atomicAdd(&cnt_sh[bi], 1.0f);     // fi: routing counts
        }

        float nrm = 1.0f / (wsum + 1e-20f);
        size_t base = (size_t)(tok0 + tid) * TOPK_K;
        float* out_w = out;
        float* out_i = out + (size_t)N * TOPK_K;
        #pragma unroll
        for (int k = 0; k < TOPK_K; ++k) {
            out_w[base + k] = tw[k] * nrm;
            out_i[base + k] = (float)ti[k];   // d_out is float; index as float value
        }
    }
    __syncthreads();

    if (tid < EXPERTS) {
        atomicAdd(&ws[tid],           pi_sh[tid]);
        atomicAdd(&ws[EXPERTS + tid], cnt_sh[tid]);
    }
}

__global__ void moe_aux_kernel(const float* __restrict__ ws,
                               float* __restrict__ out, int N)
{
    __shared__ float red[EXPERTS];
    int e = threadIdx.x;
    float pi = ws[e] / (float)N;                                    // mean prob
    float fi = ws[EXPERTS + e] * ((float)EXPERTS /
                                  ((float)N * (float)TOPK_K));      // count frac * E
    red[e] = pi * fi;
    __syncthreads();
    if (e == 0) {
        float s = 0.0f;
        for (int i = 0; i < EXPERTS; ++i) s += red[i];
        out[(size_t)2 * N * TOPK_K] = s * 0.01f;                    // aux_loss
    }
}

extern "C" void kernel_launch(void* const* d_in, const int* in_sizes, int n_in,
                              void* d_out, int out_size, void* d_ws, size_t ws_size,
                              hipStream_t stream) {
    const float* x = (const float*)d_in[0];
    const float* w = (const float*)d_in[1];
    float* out = (float*)d_out;
    float* ws  = (float*)d_ws;

    const int D = in_sizes[1] / EXPERTS;   // 2048
    const int N = in_sizes[0] / D;         // 16384

    hipMemsetAsync(ws, 0, 2 * EXPERTS * sizeof(float), stream);

    dim3 grid(N / TM), block(128);
    moe_gate_kernel<<<grid, block, 0, stream>>>(x, w, out, ws, D, N);
    moe_aux_kernel<<<1, EXPERTS, 0, stream>>>(ws, out, N);
}